// LinearCRF_28836410425741
// MI455X (gfx1250) — compile-verified
//
#include <hip/hip_runtime.h>
#include <math.h>

#define T_LEN 256
#define BATCH 256
#define HID   512
#define K_TAGS 64

typedef float v2f __attribute__((ext_vector_type(2)));
typedef float v8f __attribute__((ext_vector_type(8)));

// ---------------------------------------------------------------------------
// Kernel 1: emissions em[row=t*B+b][k] = hiddens[row][:] . W[k][:] + bias[k]
// One wave computes a 16-row x 64-col output stripe with fp32 WMMA 16x16x4.
// hiddens is read exactly once (memory-bound: 134MB @ 23.3TB/s ~ 6us).
// ---------------------------------------------------------------------------
__global__ __launch_bounds__(256) void emission_gemm(
    const float* __restrict__ hiddens,   // [T*B, H]
    const float* __restrict__ W,         // [K, H]
    const float* __restrict__ bias,      // [K]
    float* __restrict__ em)              // [T*B, K]
{
  const int wave    = (blockIdx.x * blockDim.x + threadIdx.x) >> 5; // 0..4095
  const int lane    = threadIdx.x & 31;
  const int rowBase = wave * 16;          // 16-row M tile over T*B = 65536
  const int mrow    = lane & 15;          // M (for A) / N (for B) within tile
  const int khalf   = (lane >> 4) << 1;   // K offset: 0 (lanes 0-15) or 2

  const float* aptr = hiddens + (size_t)(rowBase + mrow) * HID + khalf;

  v8f c[4];
  #pragma unroll
  for (int n = 0; n < 4; ++n) {
    const float bv = bias[n * 16 + mrow];   // bias depends only on column
    #pragma unroll
    for (int v = 0; v < 8; ++v) c[n][v] = bv;
  }

  for (int h0 = 0; h0 < HID; h0 += 4) {
    // A 16x4 fp32 tile: lane m, VGPR{0,1} = A[m][h0+khalf+{0,1}]
    v2f a = *(const v2f*)(aptr + h0);
    #pragma unroll
    for (int n = 0; n < 4; ++n) {
      // B 4x16 fp32 tile: lane n, VGPR{0,1} = W[col][h0+khalf+{0,1}]
      v2f bm = *(const v2f*)(W + (size_t)(n * 16 + mrow) * HID + h0 + khalf);
      c[n] = __builtin_amdgcn_wmma_f32_16x16x4_f32(
          /*neg_a=*/false, a, /*neg_b=*/false, bm,
          /*c_mod=*/(short)0, c[n], /*reuse_a=*/false, /*reuse_b=*/false);
    }
  }

  const int rowOff = (lane >> 4) << 3;   // C/D: lanes 16-31 hold rows M+8
  #pragma unroll
  for (int n = 0; n < 4; ++n) {
    #pragma unroll
    for (int v = 0; v < 8; ++v) {
      const int row = rowBase + rowOff + v;
      em[(size_t)row * K_TAGS + n * 16 + mrow] = c[n][v];
    }
  }
}

// ---------------------------------------------------------------------------
// Kernel 2: CRF forward (partition) scores.
// 64 threads (2 waves) per batch element; thread j owns alpha[j].
// Exp-domain matvec against exp(transition) staged transposed in LDS
// (expTc[i*64+j] -> per-wave reads hit distinct banks).
// ---------------------------------------------------------------------------
__device__ __forceinline__ float wave_max32(float v) {
  #pragma unroll
  for (int m = 16; m >= 1; m >>= 1) v = fmaxf(v, __shfl_xor(v, m, 32));
  return v;
}
__device__ __forceinline__ float wave_sum32(float v) {
  #pragma unroll
  for (int m = 16; m >= 1; m >>= 1) v += __shfl_xor(v, m, 32);
  return v;
}

__global__ __launch_bounds__(256) void crf_forward(
    const float* __restrict__ em,        // [T, B, K]
    const int*   __restrict__ lens,      // [B]
    const float* __restrict__ beginT,    // [K]
    const float* __restrict__ trans,     // [K, K]  (trans[j][i])
    const float* __restrict__ endT,      // [K]
    float* __restrict__ fs)              // [B] forward scores
{
  __shared__ float expTc[K_TAGS * K_TAGS];  // [i][j] = exp(trans[j][i])
  __shared__ float ea[4 * K_TAGS];          // exp(alpha - m) per batch group
  __shared__ float red[4 * 2];              // cross-wave reduction buffer

  const int tid  = threadIdx.x;
  for (int idx = tid; idx < K_TAGS * K_TAGS; idx += 256) {
    const int j = idx >> 6, i = idx & 63;
    expTc[i * K_TAGS + j] = __expf(trans[idx]);
  }

  const int bl   = tid >> 6;        // batch group within block (0..3)
  const int j    = tid & 63;        // tag index
  const int lane = tid & 31;
  const int half = (tid >> 5) & 1;  // which of the 2 waves of this group
  const int b    = blockIdx.x * 4 + bl;
  const int lenm1 = lens[b] - 1;

  __syncthreads();

  float alpha  = em[(size_t)b * K_TAGS + j] + beginT[j];
  float finalA = alpha;                       // correct when lenm1 == 0

  for (int t = 1; t < T_LEN; ++t) {
    // m = max_i alpha[i] over the 64 threads of this batch group
    float wm = wave_max32(alpha);
    if (lane == 0) red[bl * 2 + half] = wm;
    __syncthreads();
    const float m = fmaxf(red[bl * 2], red[bl * 2 + 1]);

    ea[bl * 64 + j] = __expf(alpha - m);
    __syncthreads();

    float s = 0.f;
    const float* eap = &ea[bl * 64];
    #pragma unroll 8
    for (int i = 0; i < K_TAGS; ++i)
      s = __fmaf_rn(eap[i], expTc[i * K_TAGS + j], s);

    alpha = m + __logf(s) + em[((size_t)t * BATCH + b) * K_TAGS + j];
    if (t == lenm1) finalA = alpha;
    __syncthreads();                          // protect ea/red for next step
  }

  // forward_score[b] = logsumexp_j(finalA[j] + endT[j])
  const float v = finalA + endT[j];
  float wm = wave_max32(v);
  if (lane == 0) red[bl * 2 + half] = wm;
  __syncthreads();
  const float m = fmaxf(red[bl * 2], red[bl * 2 + 1]);
  const float wsum = wave_sum32(__expf(v - m));
  __syncthreads();
  if (lane == 0) red[bl * 2 + half] = wsum;
  __syncthreads();
  if (j == 0) fs[b] = m + __logf(red[bl * 2] + red[bl * 2 + 1]);
}

// ---------------------------------------------------------------------------
// Kernel 3: gold path score per batch + deterministic tree reduction.
// ---------------------------------------------------------------------------
__global__ __launch_bounds__(256) void gold_and_reduce(
    const float* __restrict__ em,
    const int*   __restrict__ lens,
    const int*   __restrict__ tags,      // [T, B]
    const float* __restrict__ beginT,
    const float* __restrict__ trans,
    const float* __restrict__ endT,
    const float* __restrict__ fs,
    float* __restrict__ out)
{
  __shared__ float r[BATCH];
  const int b   = threadIdx.x;
  const int len = lens[b];

  int prev = tags[b];                               // tags[0][b]
  float g  = beginT[prev] + em[(size_t)b * K_TAGS + prev];
  for (int t = 1; t < len; ++t) {
    const int tg = tags[t * BATCH + b];
    g += trans[tg * K_TAGS + prev] + em[((size_t)t * BATCH + b) * K_TAGS + tg];
    prev = tg;
  }
  g += endT[prev];

  r[b] = fs[b] - g;
  __syncthreads();
  #pragma unroll
  for (int s2 = 128; s2 > 0; s2 >>= 1) {
    if (b < s2) r[b] += r[b + s2];
    __syncthreads();
  }
  if (b == 0) out[0] = r[0];
}

// ---------------------------------------------------------------------------
extern "C" void kernel_launch(void* const* d_in, const int* in_sizes, int n_in,
                              void* d_out, int out_size, void* d_ws, size_t ws_size,
                              hipStream_t stream) {
  const float* hiddens = (const float*)d_in[0];   // [T,B,H] f32
  const int*   lens    = (const int*)  d_in[1];   // [B]
  const int*   tags    = (const int*)  d_in[2];   // [T,B]
  const float* W       = (const float*)d_in[3];   // [K,H]
  const float* bias    = (const float*)d_in[4];   // [K]
  const float* beginT  = (const float*)d_in[5];   // [K]
  const float* trans   = (const float*)d_in[6];   // [K,K]
  const float* endT    = (const float*)d_in[7];   // [K]
  float* out = (float*)d_out;

  float* em = (float*)d_ws;                                // 16.78 MB
  float* fs = em + (size_t)T_LEN * BATCH * K_TAGS;         // + 1 KB

  // 4096 M-tiles of 16 rows, 8 waves per 256-thread block -> 512 blocks
  emission_gemm<<<512, 256, 0, stream>>>(hiddens, W, bias, em);
  crf_forward<<<BATCH / 4, 256, 0, stream>>>(em, lens, beginT, trans, endT, fs);
  gold_and_reduce<<<1, 256, 0, stream>>>(em, lens, tags, beginT, trans, endT, fs, out);
}